// mIoU_31336081391705
// MI455X (gfx1250) — compile-verified
//
#include <hip/hip_runtime.h>

// mIoU loss: mean over 4M box pairs of (1 - IoU). Memory-bound streaming
// reduction: 128 MB in, 4 B out. Strategy: B128 non-temporal loads, per-lane
// f32 accumulation, exact wave reduction via V_WMMA_F32_16X16X4_F32 (A = lane
// partials, B = ones), deterministic two-pass block/grid reduction.

typedef float v2f __attribute__((ext_vector_type(2)));
typedef float v4f __attribute__((ext_vector_type(4)));
typedef float v8f __attribute__((ext_vector_type(8)));

// Exact f32 sum of (acc0 + acc1) across all 32 lanes of the wave using the
// f32 WMMA: A is 16x4 f32 (2 VGPRs/lane = all 64 partials), B = all-ones.
// D[i][j] = rowsum(i); lanes 0-15 hold rows 0-7 in their 8 C VGPRs, lanes
// 16-31 hold rows 8-15, so one shfl_xor(16) completes the total.
// Must be called with full EXEC (all lanes converged).
__device__ __forceinline__ float wave_reduce_wmma(float acc0, float acc1) {
    v2f a; a.x = acc0; a.y = acc1;     // A matrix fragment (16x4 f32)
    v2f b; b.x = 1.0f; b.y = 1.0f;     // B matrix fragment (4x16 ones)
    v8f c = {};
    c = __builtin_amdgcn_wmma_f32_16x16x4_f32(
        /*neg_a=*/false, a, /*neg_b=*/false, b,
        /*c_mod=*/(short)0, c, /*reuse_a=*/false, /*reuse_b=*/false);
    float s = c[0] + c[1] + c[2] + c[3] + c[4] + c[5] + c[6] + c[7];
    s += __shfl_xor(s, 16, 32);        // fold other half of the rows
    return s;
}

// Block reduction: WMMA wave reduce, then LDS combine of the 8 wave sums.
// Result valid in thread 0 only.
__device__ __forceinline__ float block_reduce(float acc0, float acc1) {
    __shared__ float wsum[8];
    const int lane = threadIdx.x & 31;
    const int wave = threadIdx.x >> 5;
    float s = wave_reduce_wmma(acc0, acc1);
    if (lane == 0) wsum[wave] = s;
    __syncthreads();
    float total = 0.0f;
    if (threadIdx.x == 0) {
        const int nw = blockDim.x >> 5;
        for (int w = 0; w < nw; ++w) total += wsum[w];
    }
    return total;
}

__device__ __forceinline__ float pair_loss(const v4f o, const v4f t) {
    float area_o = (o.z - o.x) * (o.w - o.y);
    float area_t = (t.z - t.x) * (t.w - t.y);
    float iw = fminf(o.z, t.z) - fmaxf(o.x, t.x);
    float ih = fminf(o.w, t.w) - fmaxf(o.y, t.y);
    iw = fmaxf(iw, 0.0f);
    ih = fmaxf(ih, 0.0f);
    float inter = iw * ih;
    float uni   = area_o + area_t - inter;
    return 1.0f - inter / uni;
}

__global__ __launch_bounds__(256) void iou_partial_kernel(
    const v4f* __restrict__ obox, const v4f* __restrict__ tbox,
    float* __restrict__ partials, int n) {
    float acc0 = 0.0f, acc1 = 0.0f;
    const int stride = gridDim.x * blockDim.x;
    for (int i = blockIdx.x * blockDim.x + threadIdx.x; i < n; i += 2 * stride) {
        v4f o = __builtin_nontemporal_load(&obox[i]);   // global_load_b128, TH=NT
        v4f t = __builtin_nontemporal_load(&tbox[i]);
        acc0 += pair_loss(o, t);
        const int j = i + stride;
        if (j < n) {
            v4f o2 = __builtin_nontemporal_load(&obox[j]);
            v4f t2 = __builtin_nontemporal_load(&tbox[j]);
            acc1 += pair_loss(o2, t2);
        }
    }
    // All lanes reconverged here: full EXEC for the WMMA reduction.
    float total = block_reduce(acc0, acc1);
    if (threadIdx.x == 0) partials[blockIdx.x] = total;   // every block writes
}

__global__ __launch_bounds__(256) void iou_finalize_kernel(
    const float* __restrict__ partials, int nparts, float inv_n,
    float* __restrict__ out) {
    float acc0 = 0.0f, acc1 = 0.0f;
    for (int i = threadIdx.x; i < nparts; i += 512) {
        acc0 += partials[i];
        const int j = i + 256;
        if (j < nparts) acc1 += partials[j];
    }
    float total = block_reduce(acc0, acc1);
    if (threadIdx.x == 0) out[0] = total * inv_n;
}

extern "C" void kernel_launch(void* const* d_in, const int* in_sizes, int n_in,
                              void* d_out, int out_size, void* d_ws, size_t ws_size,
                              hipStream_t stream) {
    (void)n_in; (void)out_size; (void)ws_size;
    const v4f* obox = (const v4f*)d_in[0];   // output boxes, [B,4] f32
    const v4f* tbox = (const v4f*)d_in[1];   // target boxes, [B,4] f32
    const int n = in_sizes[0] / 4;           // number of box pairs

    const int THREADS = 256;
    const int BLOCKS  = 1024;                // 262k threads, ~15 pairs/thread
    float* partials = (float*)d_ws;          // BLOCKS floats of scratch (4 KB)

    iou_partial_kernel<<<BLOCKS, THREADS, 0, stream>>>(obox, tbox, partials, n);
    iou_finalize_kernel<<<1, THREADS, 0, stream>>>(partials, BLOCKS,
                                                   1.0f / (float)n,
                                                   (float*)d_out);
}